// lossFunc_6966436954242
// MI455X (gfx1250) — compile-verified
//
#include <hip/hip_runtime.h>
#include <hip/hip_bf16.h>

typedef __attribute__((ext_vector_type(2))) float v2f;
typedef __attribute__((ext_vector_type(8))) float v8f;

#define BS   2048
#define TS   200
#define QS   128
#define TM1  199                 // T-1
#define NROWS (BS * TM1)         // 407552, exact multiple of 16

// ---------------------------------------------------------------------------
// Kernel 1: streaming pass over pred + batch.
// One wave handles 16 consecutive flattened (b,t) rows.
//   A (16x4 f32 chunk)  = pred[b, t, k0..k0+3]        rows
//   B (4x16 f32 chunk)  = delta[b, t+1, k0..k0+3]     columns (same row set)
//   D = A*B + C accumulated over K=128  ->  diag(D) = row-wise dot products.
// Per-lane f32 A layout (ISA 7.12.2): lane L holds M=L%16, K={0,1} (L<16) or
// K={2,3} (L>=16) in its 2 VGPRs; B mirrors with N=L%16. Both operands index
// rows by L%16, so one pointer pair per lane serves A and B.
// Alongside, accumulate sum_q(corr - incorr) for the ground-truth bit.
// ---------------------------------------------------------------------------
__global__ __launch_bounds__(256) void k1_dot(const float* __restrict__ pred,
                                              const float* __restrict__ batch,
                                              float* __restrict__ p_raw,
                                              float* __restrict__ a_raw) {
    const int lane  = threadIdx.x & 31;
    const int wave  = threadIdx.x >> 5;
    const int group = blockIdx.x * 8 + wave;          // 16 rows per group
    const int m     = lane & 15;
    const int i     = group * 16 + m;                 // flattened (b, t) row
    const int b     = i / TM1;
    const int t     = i - b * TM1;                    // 0..198

    const float* predRow = pred  + ((size_t)b * TS + t)     * QS;
    const float* corrRow = batch + ((size_t)b * TS + t + 1) * (2 * QS);
    const float* incRow  = corrRow + QS;
    const int khalf = (lane < 16) ? 0 : 2;

    v8f   c     = {};
    float diffp = 0.0f;

#pragma unroll 8
    for (int k0 = 0; k0 < QS; k0 += 4) {
        v2f av = *(const v2f*)(predRow + k0 + khalf);
        v2f cv = *(const v2f*)(corrRow + k0 + khalf);
        v2f iv = *(const v2f*)(incRow  + k0 + khalf);
        v2f bv = cv + iv;                              // delta = corr + incorr
        diffp += (cv[0] - iv[0]) + (cv[1] - iv[1]);    // for ground-truth a
        // 8 args: (neg_a, A, neg_b, B, c_mod, C, reuse_a, reuse_b)
        c = __builtin_amdgcn_wmma_f32_16x16x4_f32(
                false, av, false, bv, (short)0, c, false, false);
    }

    // Combine the two K-half partial diff sums (lanes L and L+16 share row L%16).
    float diffo = __shfl(diffp, lane + 16, 32);
    float diff  = diffp + diffo;                       // valid for lanes 0..15

    // Extract diag(D): element (j,j) lives in VGPR (j&7), lane j (j<8) or j+16.
    const int src = lane + ((lane & 8) ? 16 : 0);
    float pdiag = 0.0f;
#pragma unroll
    for (int r = 0; r < 8; ++r) {
        float tv = __shfl(c[r], src, 32);
        if ((lane & 7) == r) pdiag = tv;
    }

    if (lane < 16) {
        p_raw[(size_t)group * 16 + lane] = pdiag;
        // a = floor((diff + 1) / 2), diff in {-1, 0, +1}
        a_raw[(size_t)group * 16 + lane] = floorf((diff + 1.0f) * 0.5f);
    }
}

// ---------------------------------------------------------------------------
// Kernel 2: one wave per student. Trailing-trim mask = (t <= last t with p>0),
// masked BCE, per-student mean -> ws; mask p/a in place in d_out.
// Butterfly shuffles give a fixed reduction order (deterministic).
// ---------------------------------------------------------------------------
__global__ __launch_bounds__(256) void k2_mask(float* __restrict__ p_io,
                                               float* __restrict__ a_io,
                                               float* __restrict__ studentLoss) {
    const int lane = threadIdx.x & 31;
    const int wave = threadIdx.x >> 5;
    const int b    = blockIdx.x * 8 + wave;           // grid 256 * 8 = 2048

    float pv[7], av[7];
    int lastp = -1;
#pragma unroll
    for (int j = 0; j < 7; ++j) {
        const int t = lane + j * 32;
        if (t < TM1) {
            pv[j] = p_io[(size_t)b * TM1 + t];
            av[j] = a_io[(size_t)b * TM1 + t];
            if (pv[j] > 0.0f) lastp = t;
        } else {
            pv[j] = 0.0f; av[j] = 0.0f;
        }
    }
    // wave-wide max of last positive index
#pragma unroll
    for (int off = 16; off; off >>= 1)
        lastp = max(lastp, __shfl_xor(lastp, off, 32));

    float bsum = 0.0f;
    int   cnt  = 0;
#pragma unroll
    for (int j = 0; j < 7; ++j) {
        const int t  = lane + j * 32;
        const bool mk = (t < TM1) && (t <= lastp);
        const float ps  = mk ? pv[j] : 0.5f;
        const float bce = -(av[j] * logf(ps) + (1.0f - av[j]) * logf(1.0f - ps));
        if (mk) { bsum += bce; cnt += 1; }
        if (t < TM1) {
            p_io[(size_t)b * TM1 + t] = mk ? pv[j] : 0.0f;
            a_io[(size_t)b * TM1 + t] = mk ? av[j] : 0.0f;
        }
    }
#pragma unroll
    for (int off = 16; off; off >>= 1) {
        bsum += __shfl_xor(bsum, off, 32);
        cnt  += __shfl_xor(cnt,  off, 32);
    }
    if (lane == 0)
        studentLoss[b] = bsum / (float)max(cnt, 1);
}

// ---------------------------------------------------------------------------
// Kernel 3: 2048 per-student losses -> scalar, fixed-shape LDS tree.
// ---------------------------------------------------------------------------
__global__ __launch_bounds__(256) void k3_reduce(const float* __restrict__ studentLoss,
                                                 float* __restrict__ out) {
    __shared__ float sm[256];
    float s = 0.0f;
#pragma unroll
    for (int j = 0; j < 8; ++j)
        s += studentLoss[threadIdx.x + j * 256];
    sm[threadIdx.x] = s;
    __syncthreads();
    for (int off = 128; off; off >>= 1) {
        if ((int)threadIdx.x < off) sm[threadIdx.x] += sm[threadIdx.x + off];
        __syncthreads();
    }
    if (threadIdx.x == 0) out[0] = sm[0];
}

extern "C" void kernel_launch(void* const* d_in, const int* in_sizes, int n_in,
                              void* d_out, int out_size, void* d_ws, size_t ws_size,
                              hipStream_t stream) {
    const float* pred  = (const float*)d_in[0];   // [B, T, Q] f32
    const float* batch = (const float*)d_in[1];   // [B, T, 2Q] f32
    float* out   = (float*)d_out;                 // [1 + NROWS + NROWS]
    float* p_raw = out + 1;                       // prediction slot
    float* a_raw = out + 1 + (size_t)NROWS;       // ground-truth slot
    float* sloss = (float*)d_ws;                  // 2048 floats

    // 407552 rows / 16 per wave / 8 waves per block = 3184 blocks
    k1_dot  <<<NROWS / 16 / 8, 256, 0, stream>>>(pred, batch, p_raw, a_raw);
    k2_mask <<<BS / 8,          256, 0, stream>>>(p_raw, a_raw, sloss);
    k3_reduce<<<1,              256, 0, stream>>>(sloss, out);
}